// GMF_90177133347635
// MI455X (gfx1250) — compile-verified
//
#include <hip/hip_runtime.h>
#include <stdint.h>

// GMF friend-attention gather kernel for gfx1250 (MI455X).
// Gather-bound: ~210MB of L2-resident row gathers, ~0.2 GFLOP. Uses the CDNA5
// async global->LDS path (GLOBAL_LOAD_ASYNC_TO_LDS_B64 + s_wait_asynccnt) to
// keep all 50 row gathers per wave in flight. Gate dots computed with a
// 32-row register-transpose butterfly reduce (31 shuffles / 32 rows, one
// vectorized sigmoid per 32 rows); coalesced writeback of gates.

#define PAD_IDX 100000
#define CLIP    50
#define DIM     64
#define WPB     4   // waves per block (128 threads)

__device__ __forceinline__ float wave_reduce_add(float v) {
  v += __shfl_xor(v, 16, 32);
  v += __shfl_xor(v, 8, 32);
  v += __shfl_xor(v, 4, 32);
  v += __shfl_xor(v, 2, 32);
  v += __shfl_xor(v, 1, 32);
  return v;
}

// Async DMA of 8 bytes/lane from global (SGPR base + VGPR offset, GVS mode)
// into LDS at per-lane offset lds_off. Tracked by ASYNCcnt.
__device__ __forceinline__ void async_row_gather(uint32_t lds_off,
                                                 uint32_t goff,
                                                 uint64_t gbase) {
  asm volatile("global_load_async_to_lds_b64 %0, %1, %2"
               :: "v"(lds_off), "v"(goff), "s"(gbase)
               : "memory");
}

// Butterfly transpose-reduce: input p[c] = this lane's partial for row c.
// Output: p[0] holds, in lane l, the full 32-lane sum for row l.
__device__ __forceinline__ void transpose_reduce32(float p[32], int lane) {
#pragma unroll
  for (int s = 16; s >= 1; s >>= 1) {
#pragma unroll
    for (int i = 0; i < s; ++i) {
      const float a = p[i];
      const float b = p[i + s];
      float t = (lane & s) ? a : b;     // send partner's row
      t = __shfl_xor(t, s, 32);
      p[i] = ((lane & s) ? b : a) + t;  // keep own row, add partner's half
    }
  }
}

__global__ __launch_bounds__(WPB * 32) void gmf_friend_attn_kernel(
    const int*   __restrict__ uidx,   // [B]
    const int*   __restrict__ iidx,   // [B]
    const int*   __restrict__ ufi,    // [NU, CLIP]
    const float* __restrict__ eu,     // [PAD_IDX+1, DIM] (pad row == 0)
    const float* __restrict__ ei,     // [NI, DIM]
    const float* __restrict__ w,      // [DIM]
    const float* __restrict__ bptr,   // scalar
    float*       __restrict__ rating, // [B]
    float*       __restrict__ gout,   // [B, CLIP]
    int B) {
  __shared__ float2 sh[WPB][CLIP][32];   // 51.2 KB: per-wave 12.8KB gather stage

  const int lane = threadIdx.x & 31;
  const int wid  = threadIdx.x >> 5;
  const int b    = blockIdx.x * WPB + wid;
  if (b >= B) return;

  const int   u    = uidx[b];
  const int   it   = iidx[b];
  const float bias = *bptr;

  // Each lane owns dims (2*lane, 2*lane+1). iw = item_emb * affine_w.
  const float2 ie = ((const float2*)(ei + (size_t)it * DIM))[lane];
  const float2 w2 = ((const float2*)w)[lane];
  const float iwx = ie.x * w2.x;
  const float iwy = ie.y * w2.y;

  // Cooperative load of the 50 friend ids; constant-lane shuffles below
  // lower to v_readlane after unroll.
  const int* frow = ufi + (size_t)u * CLIP;
  const int id_lo = frow[lane];                                     // c = 0..31
  const int id_hi = (lane < CLIP - 32) ? frow[32 + lane] : PAD_IDX; // c = 32..49

  // Low 32 bits of a generic pointer to __shared__ == LDS byte offset
  // (ISA 10.2: LDS aperture address low dword is LDS_ADDR).
  const uint32_t lds_base =
      (uint32_t)(uintptr_t)(&sh[wid][0][0]) + (uint32_t)lane * 8u;
  const uint64_t eu_base = (uint64_t)(uintptr_t)eu;

  // Issue all CLIP row gathers asynchronously (<= 63 outstanding allowed).
  int cnt = 0;
#pragma unroll
  for (int c = 0; c < CLIP; ++c) {
    const int fid = (c < 32) ? __shfl(id_lo, c, 32) : __shfl(id_hi, c - 32, 32);
    cnt += (fid != PAD_IDX);
    const uint32_t goff = (uint32_t)fid * (DIM * 4) + (uint32_t)lane * 8u;
    async_row_gather(lds_base + (uint32_t)c * (DIM * 4), goff, eu_base);
  }

  // ---- Rows 0..31: wait until first 32 gathers done (<=18 outstanding). ----
  asm volatile("s_wait_asynccnt 18" ::: "memory");
  float p[32];
#pragma unroll
  for (int c = 0; c < 32; ++c) {
    const float2 v = sh[wid][c][lane];
    p[c] = v.x * iwx + v.y * iwy;
  }
  transpose_reduce32(p, lane);
  // lane l now holds dot(row l); one vectorized sigmoid covers 32 rows
  const float g_lo = 1.0f / (1.0f + __expf(-(p[0] + bias)));

  // ---- Rows 32..49 (zero-padded tree). Wait for the remaining 18. ----
  asm volatile("s_wait_asynccnt 0" ::: "memory");
  float q[32];
#pragma unroll
  for (int c = 0; c < CLIP - 32; ++c) {
    const float2 v = sh[wid][32 + c][lane];
    q[c] = v.x * iwx + v.y * iwy;
  }
#pragma unroll
  for (int c = CLIP - 32; c < 32; ++c) q[c] = 0.0f;
  transpose_reduce32(q, lane);
  const float g_hi = 1.0f / (1.0f + __expf(-(q[0] + bias)));  // valid lanes 0..17

  // Coalesced gate writeback: lane l owns column l / 32+l directly.
  float* grow = gout + (size_t)b * CLIP;
  grow[lane] = g_lo;
  if (lane < CLIP - 32) grow[32 + lane] = g_hi;

  // ---- Accumulate user_emb = sum_c g_c * fe[c,:] (gates via v_readlane). ----
  float accx = 0.f, accy = 0.f;
#pragma unroll
  for (int c = 0; c < 32; ++c) {
    const float g = __shfl(g_lo, c, 32);
    const float2 v = sh[wid][c][lane];
    accx += v.x * g;
    accy += v.y * g;
  }
#pragma unroll
  for (int c = 0; c < CLIP - 32; ++c) {
    const float g = __shfl(g_hi, c, 32);
    const float2 v = sh[wid][32 + c][lane];
    accx += v.x * g;
    accy += v.y * g;
  }

  // user_emb = acc / friend_num; rating = sigmoid(sum(user_emb * iw) + b)
  const float fn  = (float)cnt;
  const float uex = accx / fn;
  const float uey = accy / fn;
  const float logit = wave_reduce_add(uex * iwx + uey * iwy) + bias;
  if (lane == 0) rating[b] = 1.0f / (1.0f + __expf(-logit));
}

extern "C" void kernel_launch(void* const* d_in, const int* in_sizes, int n_in,
                              void* d_out, int out_size, void* d_ws, size_t ws_size,
                              hipStream_t stream) {
  const int*   uidx = (const int*)d_in[0];
  const int*   iidx = (const int*)d_in[1];
  const int*   ufi  = (const int*)d_in[2];
  const float* eu   = (const float*)d_in[3];
  const float* ei   = (const float*)d_in[4];
  const float* w    = (const float*)d_in[5];
  const float* bb   = (const float*)d_in[6];

  const int B = in_sizes[0];
  float* rating = (float*)d_out;        // [B]
  float* gout   = (float*)d_out + B;    // [B*CLIP], concatenated after rating

  const int blocks = (B + WPB - 1) / WPB;
  gmf_friend_attn_kernel<<<blocks, WPB * 32, 0, stream>>>(
      uidx, iidx, ufi, eu, ei, w, bb, rating, gout, B);
}